// SAGELayer_47553877901461
// MI455X (gfx1250) — compile-verified
//
#include <hip/hip_runtime.h>
#include <hip/hip_bf16.h>

#define D 128

typedef __attribute__((ext_vector_type(2))) float v2f;
typedef __attribute__((ext_vector_type(8))) float v8f;

// ---------------------------------------------------------------------------
// Small utility kernels
// ---------------------------------------------------------------------------
__global__ void sage_zero_f4_kernel(float4* __restrict__ p, int n4) {
    int i = blockIdx.x * blockDim.x + threadIdx.x;
    if (i < n4) p[i] = make_float4(0.f, 0.f, 0.f, 0.f);
}

__global__ void sage_zero_i32_kernel(int* __restrict__ p, int n) {
    int i = blockIdx.x * blockDim.x + threadIdx.x;
    if (i < n) p[i] = 0;
}

// ---------------------------------------------------------------------------
// CSR build step 1: histogram of destination degrees (int atomics, cheap).
// ---------------------------------------------------------------------------
__global__ void sage_count_kernel(const int* __restrict__ dst,
                                  int* __restrict__ deg, int E) {
    int e = blockIdx.x * blockDim.x + threadIdx.x;
    if (e < E) atomicAdd(&deg[dst[e]], 1);
}

// ---------------------------------------------------------------------------
// CSR build step 2: two-level exclusive scan of deg -> row_start.
// scan1: per-block (1024) exclusive scan + block total.
// ---------------------------------------------------------------------------
__global__ __launch_bounds__(1024) void sage_scan1_kernel(
        const int* __restrict__ deg, int* __restrict__ row_start,
        int* __restrict__ block_sums, int n) {
    __shared__ int s[1024];
    const int tid = threadIdx.x;
    const int gid = blockIdx.x * 1024 + tid;
    const int v = (gid < n) ? deg[gid] : 0;
    s[tid] = v;
    __syncthreads();
#pragma unroll
    for (int off = 1; off < 1024; off <<= 1) {
        int t = (tid >= off) ? s[tid - off] : 0;
        __syncthreads();
        s[tid] += t;
        __syncthreads();
    }
    if (gid < n) row_start[gid] = s[tid] - v;      // exclusive
    if (tid == 1023) block_sums[blockIdx.x] = s[1023];
}

// scan2: serial exclusive scan of the (few) block sums.
__global__ void sage_scan2_kernel(int* __restrict__ block_sums, int nb) {
    if (blockIdx.x == 0 && threadIdx.x == 0) {
        int run = 0;
        for (int i = 0; i < nb; ++i) { int t = block_sums[i]; block_sums[i] = run; run += t; }
    }
}

// scan3: add block offsets; duplicate into the fill cursor.
__global__ __launch_bounds__(1024) void sage_scan3_kernel(
        int* __restrict__ row_start, const int* __restrict__ block_sums,
        int* __restrict__ cursor, int n) {
    int gid = blockIdx.x * 1024 + threadIdx.x;
    if (gid < n) {
        int v = row_start[gid] + block_sums[blockIdx.x];
        row_start[gid] = v;
        cursor[gid]    = v;
    }
}

// ---------------------------------------------------------------------------
// CSR build step 3: bucket edges by destination: pairs[pos] = (src, weight).
// One int atomic per edge to grab a slot.
// ---------------------------------------------------------------------------
__global__ void sage_fill_kernel(const int*   __restrict__ src,
                                 const int*   __restrict__ dst,
                                 const float* __restrict__ ew,
                                 int*  __restrict__ cursor,
                                 int2* __restrict__ pairs, int E) {
    int e = blockIdx.x * blockDim.x + threadIdx.x;
    if (e >= E) return;
    int pos = atomicAdd(&cursor[dst[e]], 1);
    pairs[pos] = make_int2(src[e], __float_as_int(ew[e]));
}

// ---------------------------------------------------------------------------
// Gather-reduce: one wave32 per destination node.  Lane l owns elements
// [4l, 4l+4).  Per edge: one uniform 8B pair load + one coalesced 512B row
// gather from L2-resident x, 4 FMAs into registers.  Row written ONCE —
// zero f32 atomics (vs 102.4M in the push formulation).
// ---------------------------------------------------------------------------
__global__ void sage_gather_kernel(const int2* __restrict__ pairs,
                                   const int*  __restrict__ row_start,
                                   const float* __restrict__ x,
                                   float* __restrict__ neigh,
                                   int N, int E) {
    const int node = blockIdx.x * 8 + (threadIdx.x >> 5);
    const int lane = threadIdx.x & 31;
    if (node >= N) return;

    const int start = row_start[node];
    const int end   = (node + 1 < N) ? row_start[node + 1] : E;
    const float4* x4 = (const float4*)x;

    float4 acc = make_float4(0.f, 0.f, 0.f, 0.f);
    for (int i = start; i < end; ++i) {
        const int2  p = pairs[i];
        const float w = __int_as_float(p.y);
        const float4 v = x4[(size_t)p.x * (D / 4) + lane];
        acc.x += v.x * w; acc.y += v.y * w; acc.z += v.z * w; acc.w += v.w * w;
    }
    ((float4*)neigh)[(size_t)node * (D / 4) + lane] = acc;
}

// ---------------------------------------------------------------------------
// Fallback (ws too small for CSR): direct atomic scatter, one wave per edge,
// HW global_atomic_add_f32 via unsafeAtomicAdd.
// ---------------------------------------------------------------------------
__global__ void sage_scatter_kernel(const float* __restrict__ x,
                                    const float* __restrict__ edge_w,
                                    const int*   __restrict__ src,
                                    const int*   __restrict__ dst,
                                    float*       __restrict__ neigh,
                                    int E) {
    int tid  = blockIdx.x * blockDim.x + threadIdx.x;
    int edge = tid >> 5;
    int lane = tid & 31;
    if (edge >= E) return;
    int   s = src[edge];
    int   d = dst[edge];
    float w = edge_w[edge];
    const float4 v = *(const float4*)(x + (size_t)s * D + lane * 4);
    float* base = neigh + (size_t)d * D + lane * 4;
    unsafeAtomicAdd(base + 0, v.x * w);
    unsafeAtomicAdd(base + 1, v.y * w);
    unsafeAtomicAdd(base + 2, v.z * w);
    unsafeAtomicAdd(base + 3, v.w * w);
}

// ---------------------------------------------------------------------------
// Fused dual GEMM + bias via V_WMMA_F32_16X16X4_F32 (schedule validated in
// round 3: ds_load_2addr_b64 pipelined under back-to-back WMMA pairs).
//   out = x @ W_self^T + b_self + neigh @ W_neigh^T
// ---------------------------------------------------------------------------
__global__ __launch_bounds__(160, 1) void sage_gemm_kernel(
        const float* __restrict__ x,
        const float* __restrict__ Wself,
        const float* __restrict__ bself,
        const float* __restrict__ Wneigh,
        const float* __restrict__ neigh,
        float* __restrict__ out,
        int Nrows) {
    extern __shared__ float sW[];            // [2][128*128] = 128 KB
    float* sWs = sW;
    float* sWn = sW + D * D;

    {
        const float4* gs = (const float4*)Wself;
        const float4* gn = (const float4*)Wneigh;
        float4* ss = (float4*)sWs;
        float4* sn = (float4*)sWn;
        for (int i = threadIdx.x; i < (D * D) / 4; i += blockDim.x) {
            ss[i] = gs[i];
            sn[i] = gn[i];
        }
    }
    __syncthreads();

    const int lane   = threadIdx.x & 31;
    const int wave   = threadIdx.x >> 5;
    const int m_base = (blockIdx.x * 5 + wave) * 16;

    const int r  = lane & 15;          // M row (A) or N col (B, C/D)
    const int kh = (lane >> 4) * 2;    // K sub-offset for this lane half

    v8f acc[8];
#pragma unroll
    for (int nt = 0; nt < 8; ++nt) acc[nt] = (v8f)(0.f);

    int arow = m_base + r;
    if (arow >= Nrows) arow = Nrows - 1;           // clamp (WMMA needs EXEC all-1)
    const float* xrow = x     + (size_t)arow * D;
    const float* nrow = neigh + (size_t)arow * D;

#pragma unroll 4
    for (int k0 = 0; k0 < D; k0 += 4) {
        const v2f a = *(const v2f*)(xrow + k0 + kh);
        v2f b[8];
#pragma unroll
        for (int nt = 0; nt < 8; ++nt)
            b[nt] = *(const v2f*)(sWs + (size_t)(nt * 16 + r) * D + k0 + kh);
#pragma unroll
        for (int nt = 0; nt < 8; ++nt)
            acc[nt] = __builtin_amdgcn_wmma_f32_16x16x4_f32(
                false, a, false, b[nt], (short)0, acc[nt], false, false);
    }

#pragma unroll 4
    for (int k0 = 0; k0 < D; k0 += 4) {
        const v2f a = *(const v2f*)(nrow + k0 + kh);
        v2f b[8];
#pragma unroll
        for (int nt = 0; nt < 8; ++nt)
            b[nt] = *(const v2f*)(sWn + (size_t)(nt * 16 + r) * D + k0 + kh);
#pragma unroll
        for (int nt = 0; nt < 8; ++nt)
            acc[nt] = __builtin_amdgcn_wmma_f32_16x16x4_f32(
                false, a, false, b[nt], (short)0, acc[nt], false, false);
    }

    const int row_off = 8 * (lane >> 4);
#pragma unroll
    for (int nt = 0; nt < 8; ++nt) {
        const int n    = nt * 16 + r;
        const float bv = bself[n];
#pragma unroll
        for (int i = 0; i < 8; ++i) {
            const int row = m_base + i + row_off;
            if (row < Nrows) out[(size_t)row * D + n] = acc[nt][i] + bv;
        }
    }
}

// ---------------------------------------------------------------------------
// Inputs (setup_inputs order): x, W_self, b_self, W_neigh, edge_w, src, dst
// ---------------------------------------------------------------------------
extern "C" void kernel_launch(void* const* d_in, const int* in_sizes, int n_in,
                              void* d_out, int out_size, void* d_ws, size_t ws_size,
                              hipStream_t stream) {
    const float* x      = (const float*)d_in[0];
    const float* Wself  = (const float*)d_in[1];
    const float* bself  = (const float*)d_in[2];
    const float* Wneigh = (const float*)d_in[3];
    const float* edge_w = (const float*)d_in[4];
    const int*   src    = (const int*)d_in[5];
    const int*   dst    = (const int*)d_in[6];
    float* out = (float*)d_out;

    const int N = in_sizes[0] / D;   // 50000
    const int E = in_sizes[5];       // 800000

    // Workspace layout
    char* ws = (char*)d_ws;
    float* neigh = (float*)ws;                       // N*D floats
    size_t off = (size_t)N * D * sizeof(float);
    int*  row_start = (int*)(ws + off);  off += (size_t)N * sizeof(int);
    int*  cursor    = (int*)(ws + off);  off += (size_t)N * sizeof(int);
    int2* pairs     = (int2*)(ws + off); off += (size_t)E * sizeof(int2);
    const int nb = (N + 1023) / 1024;                // scan blocks (49)
    int*  block_sums = (int*)(ws + off); off += (size_t)nb * sizeof(int);
    const bool csr_ok = (off <= ws_size);

    if (csr_ok) {
        // --- pull formulation: CSR build (int atomics) + gather-reduce ---
        sage_zero_i32_kernel<<<(N + 255) / 256, 256, 0, stream>>>(cursor, N);
        sage_count_kernel<<<(E + 255) / 256, 256, 0, stream>>>(dst, cursor, E);
        sage_scan1_kernel<<<nb, 1024, 0, stream>>>(cursor, row_start, block_sums, N);
        sage_scan2_kernel<<<1, 64, 0, stream>>>(block_sums, nb);
        sage_scan3_kernel<<<nb, 1024, 0, stream>>>(row_start, block_sums, cursor, N);
        sage_fill_kernel<<<(E + 255) / 256, 256, 0, stream>>>(src, dst, edge_w,
                                                              cursor, pairs, E);
        sage_gather_kernel<<<(N + 7) / 8, 256, 0, stream>>>(pairs, row_start,
                                                            x, neigh, N, E);
    } else {
        // --- fallback: push formulation with HW f32 atomics ---
        const int n4 = N * (D / 4);
        sage_zero_f4_kernel<<<(n4 + 255) / 256, 256, 0, stream>>>((float4*)neigh, n4);
        const long long sc_threads = (long long)E * 32;
        sage_scatter_kernel<<<(int)((sc_threads + 255) / 256), 256, 0, stream>>>(
            x, edge_w, src, dst, neigh, E);
    }

    // --- fused dual WMMA GEMM + bias (128 KB dynamic LDS) ---
    const int wave_tiles = (N + 15) / 16;            // 3125
    const int blocks     = (wave_tiles + 4) / 5;     // 625 (5 waves/block)
    const size_t lds_bytes = 2 * D * D * sizeof(float);
    sage_gemm_kernel<<<blocks, 160, lds_bytes, stream>>>(
        x, Wself, bself, Wneigh, neigh, out, N);
}